// CLR_9826885173712
// MI455X (gfx1250) — compile-verified
//
#include <hip/hip_runtime.h>
#include <hip/hip_bf16.h>
#include <math.h>

#define PS        32768
#define DIN       1024
#define DH        512
#define DATT      128
#define KMASK     16384
#define LENKEEP   16384

typedef __attribute__((ext_vector_type(2))) float v2f;
typedef __attribute__((ext_vector_type(8))) float v8f;

__device__ __forceinline__ v8f wmma_f32_4(v2f a, v2f b, v8f c) {
    // D = A(16x4,f32) * B(4x16,f32) + C(16x16,f32), exact fp32 on the matrix pipe
    return __builtin_amdgcn_wmma_f32_16x16x4_f32(false, a, false, b, (short)0, c,
                                                 false, false);
}

__device__ __forceinline__ float gelu_exact(float v) {
    return 0.5f * v * (1.0f + erff(v * 0.70710678118654752440f));
}

// CDNA5 async copy: 16B global -> LDS per lane, tracked by ASYNCcnt.
__device__ __forceinline__ void async_copy_b128(unsigned lds_off, const float* gptr) {
    asm volatile("global_load_async_to_lds_b128 %0, %1, off"
                 :: "v"(lds_off), "v"(gptr) : "memory");
}
__device__ __forceinline__ void wait_async0() {
    asm volatile("s_wait_asynccnt 0x0" ::: "memory");
}

// ---------------------------------------------------------------------------
// Pack W (K x N, row major) into WMMA-B fragment order:
// Wp4[(k8*N + n)*2 + kh] = { W[(8k8+2kh+0)N+n], W[(8k8+2kh+1)N+n],
//                            W[(8k8+2kh+4)N+n], W[(8k8+2kh+5)N+n] }
// so one b128 load per lane feeds TWO k-steps of one 16x16 tile.
// ---------------------------------------------------------------------------
__global__ __launch_bounds__(256) void pack_w_kernel(
    const float* __restrict__ W, float4* __restrict__ Wp, int K, int N)
{
    const int idx = blockIdx.x * 256 + threadIdx.x;
    const int total = (K >> 3) * N * 2;
    if (idx >= total) return;
    const int kh = idx & 1;
    const int n  = (idx >> 1) % N;
    const int k8 = (idx >> 1) / N;
    const int kg = k8 * 8 + kh * 2;
    float4 v;
    v.x = W[(size_t)(kg + 0) * N + n];
    v.y = W[(size_t)(kg + 1) * N + n];
    v.z = W[(size_t)(kg + 4) * N + n];
    v.w = W[(size_t)(kg + 5) * N + n];
    Wp[idx] = v;
}

// ---------------------------------------------------------------------------
// Kernel 1: exact top-k complement selection (single block).
// 3-round radix select on float bits -> exact threshold T and E equals masked.
// ---------------------------------------------------------------------------
__global__ __launch_bounds__(1024) void topk_keep_kernel(
    const float* __restrict__ attn, int* __restrict__ keep)
{
    __shared__ unsigned int hist[2048];
    __shared__ unsigned int tsum[1024];
    __shared__ unsigned int eqIdx[64];
    __shared__ unsigned int sh_prefix, sh_remaining, sh_eqCount;

    const int tid = threadIdx.x;
    if (tid == 0) { sh_prefix = 0u; sh_remaining = KMASK; }
    __syncthreads();

    const int rbits[3]  = {11, 11, 10};
    const int rshift[3] = {21, 10, 0};
    for (int r = 0; r < 3; ++r) {
        const int nb = 1 << rbits[r];
        for (int i = tid; i < nb; i += 1024) hist[i] = 0u;
        __syncthreads();
        const unsigned pref = sh_prefix;
        const int topshift = rshift[r] + rbits[r];
        for (int i = tid; i < PS; i += 1024) {
            unsigned key = __float_as_uint(attn[i]);
            unsigned hi  = (topshift >= 32) ? 0u : (key >> topshift);
            if (hi == pref)
                atomicAdd(&hist[(key >> rshift[r]) & (nb - 1)], 1u);
        }
        __syncthreads();
        if (tid == 0) {
            unsigned rem = sh_remaining, cum = 0; int sel = 0;
            for (int b = nb - 1; b >= 0; --b) {
                unsigned c = hist[b];
                if (cum + c >= rem) { sel = b; sh_remaining = rem - cum; break; }
                cum += c;
            }
            sh_prefix = (pref << rbits[r]) | (unsigned)sel;
        }
        __syncthreads();
    }
    const unsigned T = sh_prefix;
    const unsigned E = sh_remaining;

    if (tid == 0) {                     // lowest-index equals are masked
        unsigned cnt = 0;
        for (int i = 0; i < PS && cnt < E; ++i)
            if (__float_as_uint(attn[i]) == T) { if (cnt < 64) eqIdx[cnt] = (unsigned)i; ++cnt; }
        sh_eqCount = (cnt < 64u) ? cnt : 64u;
    }
    __syncthreads();
    const unsigned eqCount = sh_eqCount;

    const int base = tid * 32;
    int flags[32]; int cnt = 0;
    for (int j = 0; j < 32; ++j) {
        const int i = base + j;
        const unsigned key = __float_as_uint(attn[i]);
        int kp;
        if      (key < T) kp = 1;
        else if (key > T) kp = 0;
        else {
            kp = 1;
            for (unsigned e = 0; e < eqCount; ++e)
                if (eqIdx[e] == (unsigned)i) { kp = 0; break; }
        }
        flags[j] = kp; cnt += kp;
    }
    tsum[tid] = (unsigned)cnt;
    __syncthreads();
    if (tid == 0) {
        unsigned run = 0;
        for (int i = 0; i < 1024; ++i) { unsigned c = tsum[i]; tsum[i] = run; run += c; }
    }
    __syncthreads();
    unsigned pos = tsum[tid];
    for (int j = 0; j < 32; ++j)
        if (flags[j]) keep[pos++] = base + j;
}

// ---------------------------------------------------------------------------
// Kernel 2: hk = relu(x[keep,:] @ W_emb + b_emb), fp32 WMMA.
// Block = 64 gathered rows x all 512 cols; 8 waves, wave owns 64 cols.
// Per wave: 4 M-tiles x 4 N-tiles = 16 accumulators.
// Inner kk-step (K=8): 8 LDS b64 (A) + 4 global b128 (packed B) + 32 WMMA.
// A staged via global_load_async_to_lds_b128 (ASYNCcnt).
// ---------------------------------------------------------------------------
__global__ __launch_bounds__(256) void gemm_emb_kernel(
    const float* __restrict__ x, const float4* __restrict__ Wp,
    const float* __restrict__ bias, const int* __restrict__ keep,
    float* __restrict__ hk)
{
    __shared__ __align__(16) float As[64 * 128];   // 32 KB

    const int mbase = blockIdx.x * 64;
    const int tid   = threadIdx.x;
    const int lane  = tid & 31;
    const int lrow  = lane & 15;
    const int kh    = lane >> 4;                   // fragment K-half select
    const int kh2   = kh * 2;
    const int n0w   = (tid >> 5) * 64;

    // gather: 8 row pointers per thread (rows w, w+8, ..., w+56)
    const float* rowp[8];
    #pragma unroll
    for (int rep = 0; rep < 8; ++rep)
        rowp[rep] = x + (size_t)keep[mbase + (tid >> 5) + 8 * rep] * DIN;

    v8f acc[4][4];
    const v8f zero = {0.f,0.f,0.f,0.f,0.f,0.f,0.f,0.f};
    #pragma unroll
    for (int mt = 0; mt < 4; ++mt)
        #pragma unroll
        for (int nt = 0; nt < 4; ++nt) acc[mt][nt] = zero;

    const int c4 = (tid & 31) * 4;                 // float offset of this lane's b128

    for (int kc = 0; kc < DIN; kc += 128) {
        // stage A: 64 rows x 128 K = 2048 float4; async copy, 8 per thread
        #pragma unroll
        for (int rep = 0; rep < 8; ++rep) {
            const int r = (tid >> 5) + 8 * rep;
            async_copy_b128((unsigned)(uintptr_t)(As + r * 128 + c4),
                            rowp[rep] + kc + c4);
        }
        wait_async0();
        __syncthreads();

        for (int kk = 0; kk < 128; kk += 8) {
            v2f a0[4], a1[4];
            #pragma unroll
            for (int mt = 0; mt < 4; ++mt) {
                const float* ap = As + (mt * 16 + lrow) * 128 + kk + kh2;
                a0[mt] = *(const v2f*)(ap);
                a1[mt] = *(const v2f*)(ap + 4);
            }
            const int k8g = (kc + kk) >> 3;
            #pragma unroll
            for (int nt = 0; nt < 4; ++nt) {
                const int n = n0w + nt * 16 + lrow;
                const float4 bq = Wp[((size_t)k8g * DH + n) * 2 + kh];
                const v2f b0 = {bq.x, bq.y};
                const v2f b1 = {bq.z, bq.w};
                #pragma unroll
                for (int mt = 0; mt < 4; ++mt) {
                    acc[mt][nt] = wmma_f32_4(a0[mt], b0, acc[mt][nt]);
                    acc[mt][nt] = wmma_f32_4(a1[mt], b1, acc[mt][nt]);
                }
            }
        }
        __syncthreads();
    }

    // epilogue: +bias, relu.  C layout: VGPR v -> M=v (lanes<16) / v+8
    const int moff = kh * 8;
    #pragma unroll
    for (int nt = 0; nt < 4; ++nt) {
        const int n  = n0w + nt * 16 + lrow;
        const float bn = bias[n];
        #pragma unroll
        for (int mt = 0; mt < 4; ++mt) {
            #pragma unroll
            for (int v = 0; v < 8; ++v) {
                const int row = mbase + mt * 16 + v + moff;
                float val = acc[mt][nt][v] + bn;
                hk[(size_t)row * DH + n] = val > 0.f ? val : 0.f;
            }
        }
    }
}

// ---------------------------------------------------------------------------
// Kernel 3: scores = gelu(hk @ W_a1 + b_a1) @ W_a2 + b_a2.
// Block = 64 rows x 128 cols; 4 waves, wave owns 32 cols (2 N-tiles x 4 M-tiles).
// Shared buffer reused: A-tile (64x128) during GEMM, post-gelu S1 (64x136) after.
// ---------------------------------------------------------------------------
__global__ __launch_bounds__(128) void attn_score_kernel(
    const float* __restrict__ hk, const float4* __restrict__ W1p,
    const float* __restrict__ b1, const float* __restrict__ W2,
    const float* __restrict__ b2, float* __restrict__ scores)
{
    __shared__ __align__(16) float smem[64 * 136];   // ~34 KB (A-tile then S1)
    __shared__ float red[64][2];

    const int mbase = blockIdx.x * 64;
    const int tid   = threadIdx.x;
    const int lane  = tid & 31;
    const int lrow  = lane & 15;
    const int kh    = lane >> 4;
    const int kh2   = kh * 2;
    const int n0w   = (tid >> 5) * 32;
    const int c4    = (tid & 31) * 4;

    v8f acc[4][2];
    const v8f zero = {0.f,0.f,0.f,0.f,0.f,0.f,0.f,0.f};
    #pragma unroll
    for (int mt = 0; mt < 4; ++mt) { acc[mt][0] = zero; acc[mt][1] = zero; }

    for (int kc = 0; kc < DH; kc += 128) {
        // stage A: 64 rows x 128 K = 2048 float4; 16 per thread (async)
        #pragma unroll
        for (int rep = 0; rep < 16; ++rep) {
            const int r = (tid >> 5) + 4 * rep;
            async_copy_b128((unsigned)(uintptr_t)(smem + r * 128 + c4),
                            hk + (size_t)(mbase + r) * DH + kc + c4);
        }
        wait_async0();
        __syncthreads();

        for (int kk = 0; kk < 128; kk += 8) {
            v2f a0[4], a1[4];
            #pragma unroll
            for (int mt = 0; mt < 4; ++mt) {
                const float* ap = smem + (mt * 16 + lrow) * 128 + kk + kh2;
                a0[mt] = *(const v2f*)(ap);
                a1[mt] = *(const v2f*)(ap + 4);
            }
            const int k8g = (kc + kk) >> 3;
            #pragma unroll
            for (int nt = 0; nt < 2; ++nt) {
                const int n = n0w + nt * 16 + lrow;
                const float4 bq = W1p[((size_t)k8g * DATT + n) * 2 + kh];
                const v2f b0 = {bq.x, bq.y};
                const v2f b1 = {bq.z, bq.w};
                #pragma unroll
                for (int mt = 0; mt < 4; ++mt) {
                    acc[mt][nt] = wmma_f32_4(a0[mt], b0, acc[mt][nt]);
                    acc[mt][nt] = wmma_f32_4(a1[mt], b1, acc[mt][nt]);
                }
            }
        }
        __syncthreads();
    }

    // S1 = gelu(acc + b1) into smem with stride 136
    const int moff = kh * 8;
    #pragma unroll
    for (int nt = 0; nt < 2; ++nt) {
        const int n = n0w + nt * 16 + lrow;
        const float bn = b1[n];
        #pragma unroll
        for (int mt = 0; mt < 4; ++mt)
            #pragma unroll
            for (int v = 0; v < 8; ++v)
                smem[(mt * 16 + v + moff) * 136 + n] = gelu_exact(acc[mt][nt][v] + bn);
    }
    __syncthreads();

    // score[row] = sum_c S1[row][c] * W2[c] + b2
    const int row  = tid >> 1;
    const int half = tid & 1;
    float p = 0.f;
    for (int j = 0; j < 64; ++j) {
        const int c = half * 64 + j;
        p += smem[row * 136 + c] * W2[c];
    }
    red[row][half] = p;
    __syncthreads();
    if (tid < 64)
        scores[mbase + tid] = red[tid][0] + red[tid][1] + b2[0];
}

// ---------------------------------------------------------------------------
// Kernel 4: softmax stats over 16384 scores (single block): stats = {max, Z}
// ---------------------------------------------------------------------------
__global__ __launch_bounds__(1024) void softmax_stats_kernel(
    const float* __restrict__ scores, float* __restrict__ stats)
{
    __shared__ float red[1024];
    __shared__ float sh_m;
    const int tid = threadIdx.x;

    float m = -3.402823466e+38f;
    for (int i = tid; i < LENKEEP; i += 1024) m = fmaxf(m, scores[i]);
    red[tid] = m; __syncthreads();
    for (int s = 512; s > 0; s >>= 1) {
        if (tid < s) red[tid] = fmaxf(red[tid], red[tid + s]);
        __syncthreads();
    }
    if (tid == 0) sh_m = red[0];
    __syncthreads();
    const float mm = sh_m;

    float z = 0.f;
    for (int i = tid; i < LENKEEP; i += 1024) z += expf(scores[i] - mm);
    red[tid] = z; __syncthreads();
    for (int s = 512; s > 0; s >>= 1) {
        if (tid < s) red[tid] += red[tid + s];
        __syncthreads();
    }
    if (tid == 0) { stats[0] = mm; stats[1] = red[0]; }
}

// ---------------------------------------------------------------------------
// Kernel 5: feat = softmax(scores) @ hk, deterministic 2-stage reduction.
// ---------------------------------------------------------------------------
__global__ __launch_bounds__(256) void feat_partial_kernel(
    const float* __restrict__ hk, const float* __restrict__ scores,
    const float* __restrict__ stats, float* __restrict__ partials)
{
    __shared__ float wbuf[128];
    const int b   = blockIdx.x;
    const int tid = threadIdx.x;
    const float m = stats[0];
    if (tid < 128) wbuf[tid] = expf(scores[b * 128 + tid] - m);
    __syncthreads();

    const int c0 = tid, c1 = tid + 256;
    float a0 = 0.f, a1 = 0.f;
    for (int j = 0; j < 128; ++j) {
        const size_t n = (size_t)(b * 128 + j);
        const float w = wbuf[j];
        a0 += w * hk[n * DH + c0];
        a1 += w * hk[n * DH + c1];
    }
    partials[b * DH + c0] = a0;
    partials[b * DH + c1] = a1;
}

__global__ __launch_bounds__(512) void feat_reduce_kernel(
    const float* __restrict__ partials, const float* __restrict__ stats,
    float* __restrict__ feat)
{
    const int c = threadIdx.x;
    float s = 0.f;
    for (int b = 0; b < 128; ++b) s += partials[b * DH + c];
    feat[c] = s / stats[1];
}

// ---------------------------------------------------------------------------
// Kernel 6: logits + CE loss (single block, fixed-order reductions)
// ---------------------------------------------------------------------------
__device__ __forceinline__ float red512_sum(float v, float* red, int tid) {
    red[tid] = v; __syncthreads();
    for (int s = 256; s > 0; s >>= 1) {
        if (tid < s) red[tid] += red[tid + s];
        __syncthreads();
    }
    float r = red[0]; __syncthreads();
    return r;
}
__device__ __forceinline__ float red512_max(float v, float* red, int tid) {
    red[tid] = v; __syncthreads();
    for (int s = 256; s > 0; s >>= 1) {
        if (tid < s) red[tid] = fmaxf(red[tid], red[tid + s]);
        __syncthreads();
    }
    float r = red[0]; __syncthreads();
    return r;
}

__global__ __launch_bounds__(512) void finalize_kernel(
    const float* __restrict__ feat, const float* __restrict__ teacher,
    const float* __restrict__ Wp, const float* __restrict__ bp,
    float* __restrict__ out)
{
    __shared__ float red[512];
    const int c = threadIdx.x;
    const float f = feat[c];
    const float t = teacher[c];

    const float mt  = red512_max(t, red, c);
    const float Zt  = red512_sum(expf(t - mt), red, c);
    const float mf  = red512_max(f, red, c);
    const float Zf  = red512_sum(expf(f - mf), red, c);
    const float lZf = logf(Zf);
    const float pt  = expf(t - mt) / Zt;

    const float loss = red512_sum(-pt * (f - mf - lZf), red, c);
    const float l0   = red512_sum(f * Wp[c * 2 + 0], red, c);
    const float l1   = red512_sum(f * Wp[c * 2 + 1], red, c);

    if (c == 0) {
        out[0] = l0 + bp[0];
        out[1] = l1 + bp[1];
        out[2] = loss;
    }
}

// ---------------------------------------------------------------------------
extern "C" void kernel_launch(void* const* d_in, const int* in_sizes, int n_in,
                              void* d_out, int out_size, void* d_ws, size_t ws_size,
                              hipStream_t stream) {
    const float* x       = (const float*)d_in[0];
    const float* attn    = (const float*)d_in[1];
    const float* teacher = (const float*)d_in[2];
    const float* W_emb   = (const float*)d_in[3];
    const float* b_emb   = (const float*)d_in[4];
    const float* W_a1    = (const float*)d_in[5];
    const float* b_a1    = (const float*)d_in[6];
    const float* W_a2    = (const float*)d_in[7];
    const float* b_a2    = (const float*)d_in[8];
    const float* W_pred  = (const float*)d_in[9];
    const float* b_pred  = (const float*)d_in[10];
    float* out = (float*)d_out;

    char* ws = (char*)d_ws;
    int*    keep     = (int*)   (ws + 0);                //   64 KB
    float*  scores   = (float*) (ws + (64u  << 10));     //   64 KB
    float*  stats    = (float*) (ws + (128u << 10));     //    8 B
    float*  feat     = (float*) (ws + (132u << 10));     //    2 KB
    float*  partials = (float*) (ws + (144u << 10));     //  256 KB
    float4* Wp_emb   = (float4*)(ws + (512u << 10));     //    2 MB
    float4* Wp_a1    = (float4*)(ws + (2560u << 10));    //  256 KB
    float*  hk       = (float*) (ws + (4u   << 20));     //   32 MB

    // weight packing (deterministic, recomputed every call)
    pack_w_kernel<<<(DIN/8)*DH*2/256, 256, 0, stream>>>(W_emb, Wp_emb, DIN, DH);
    pack_w_kernel<<<(DH/8)*DATT*2/256, 256, 0, stream>>>(W_a1, Wp_a1, DH, DATT);

    topk_keep_kernel    <<<1,              1024, 0, stream>>>(attn, keep);
    gemm_emb_kernel     <<<LENKEEP / 64,    256, 0, stream>>>(x, Wp_emb, b_emb, keep, hk);
    attn_score_kernel   <<<LENKEEP / 64,    128, 0, stream>>>(hk, Wp_a1, b_a1, W_a2, b_a2, scores);
    softmax_stats_kernel<<<1,              1024, 0, stream>>>(scores, stats);
    feat_partial_kernel <<<128,             256, 0, stream>>>(hk, scores, stats, partials);
    feat_reduce_kernel  <<<1,               512, 0, stream>>>(partials, stats, feat);
    finalize_kernel     <<<1,               512, 0, stream>>>(feat, teacher, W_pred, b_pred, out);
}